// AttentionLayer_11055245820581
// MI455X (gfx1250) — compile-verified
//
#include <hip/hip_runtime.h>

// Bahdanau additive attention, fused, fp32, CDNA5 (gfx1250, wave32).
//   inputs : enc [B,Tx,E], dec [B,Ty,D], W_a [E,E], U_a [D,E], V_a [E,1]
//   outputs: c [B,Ty,E] ++ e [B,Ty,Tx]  (concatenated in d_out)
// B=8, Tx=Ty=E=D=256.

typedef __attribute__((ext_vector_type(2))) float v2f;
typedef __attribute__((ext_vector_type(8))) float v8f;

#define BSZ 8
#define TX  256
#define TY  256
#define EE  256

// ---------------------------------------------------------------------------
// tanh: use gfx1250 v_tanh_f32 trans op when the builtin exists.
// ---------------------------------------------------------------------------
__device__ __forceinline__ float fast_tanh(float x) {
#if defined(__AMDGCN__) && __has_builtin(__builtin_amdgcn_tanhf)
  return __builtin_amdgcn_tanhf(x);
#elif defined(__AMDGCN__) && __has_builtin(__builtin_amdgcn_tanh_f32)
  return __builtin_amdgcn_tanh_f32(x);
#else
  return tanhf(x);
#endif
}

// ---------------------------------------------------------------------------
// One wave computes C[m0:m0+16, n0:n0+64] of C = A @ B (all row-major,
// K = N = 256) with V_WMMA_F32_16X16X4_F32, reusing the A fragment across
// four N-tiles per k-step.
//
// ISA 7.12.2 layouts (wave32):
//   A 16x4 : lanes 0-15 -> M=lane, VGPR0=K0 VGPR1=K1; lanes 16-31 -> K2/K3
//   B 4x16 : lanes 0-15 -> N=lane, VGPR0=K0 VGPR1=K1; lanes 16-31 -> K2/K3
//   C 16x16: VGPR g -> M = g + 8*(lane>=16), N = lane&15
// ---------------------------------------------------------------------------
__device__ __forceinline__ void wave_gemm_tile_f32(
    const float* __restrict__ A,
    const float* __restrict__ B,
    float* __restrict__ C,
    int m0, int n0)
{
  const int lane = threadIdx.x & 31;
  const int r    = lane & 15;        // row (A) / col (B,C) within tile
  const int half = lane >> 4;
  const int kk   = half << 1;        // K sub-offset for this lane half

  v8f acc0 = {}, acc1 = {}, acc2 = {}, acc3 = {};
  const float* arow = A + (m0 + r) * EE + kk;

  for (int k = 0; k < EE; k += 4) {
    v2f a;
    a.x = arow[k];
    a.y = arow[k + 1];

    const float* bb = B + (k + kk) * EE + n0 + r;
    v2f b0, b1, b2, b3;
    b0.x = bb[0];       b0.y = bb[EE];
    b1.x = bb[16];      b1.y = bb[EE + 16];
    b2.x = bb[32];      b2.y = bb[EE + 32];
    b3.x = bb[48];      b3.y = bb[EE + 48];

    acc0 = __builtin_amdgcn_wmma_f32_16x16x4_f32(false, a, false, b0,
                                                 (short)0, acc0, false, false);
    acc1 = __builtin_amdgcn_wmma_f32_16x16x4_f32(false, a, false, b1,
                                                 (short)0, acc1, false, false);
    acc2 = __builtin_amdgcn_wmma_f32_16x16x4_f32(false, a, false, b2,
                                                 (short)0, acc2, false, false);
    acc3 = __builtin_amdgcn_wmma_f32_16x16x4_f32(false, a, false, b3,
                                                 (short)0, acc3, false, false);
  }

#pragma unroll
  for (int g = 0; g < 8; ++g) {
    const int row = m0 + g + 8 * half;
    float* crow = C + row * EE + n0 + r;
    crow[0]  = acc0[g];
    crow[16] = acc1[g];
    crow[32] = acc2[g];
    crow[48] = acc3[g];
  }
}

// ---------------------------------------------------------------------------
// Stage 1: Ws = enc @ W_a  (z==0),  Uh = dec @ U_a  (z==1).
// enc/dec flatten to [B*T, E] row-major. 4 waves/block, 64 rows x 64 cols.
// ---------------------------------------------------------------------------
__global__ __launch_bounds__(128)
void proj_kernel(const float* __restrict__ enc,
                 const float* __restrict__ dec,
                 const float* __restrict__ Wa,
                 const float* __restrict__ Ua,
                 float* __restrict__ Ws,
                 float* __restrict__ Uh)
{
  const int wave = threadIdx.x >> 5;
  const int m0 = (blockIdx.y * 4 + wave) * 16;
  const int n0 = blockIdx.x * 64;
  if (blockIdx.z == 0) {
    wave_gemm_tile_f32(enc, Wa, Ws, m0, n0);
  } else {
    wave_gemm_tile_f32(dec, Ua, Uh, m0, n0);
  }
}

// ---------------------------------------------------------------------------
// Stage 2: per (b,y): e[x] = softmax_x( sum_e tanh(Ws[b,x,e]+Uh[b,y,e])*V[e] )
// One 256-thread block per (b,y); Uh row + V staged in LDS (uniform-index
// broadcast, bank-conflict-free); each thread streams its Ws row as float4.
// ---------------------------------------------------------------------------
__global__ __launch_bounds__(256)
void score_kernel(const float* __restrict__ Ws,
                  const float* __restrict__ Uh,
                  const float* __restrict__ Va,
                  float* __restrict__ e_out)
{
  __shared__ float uh[EE];
  __shared__ float vv[EE];
  __shared__ float red[256];

  const int tid = threadIdx.x;
  const int by  = blockIdx.x;        // flattened (b,y), b = by >> 8
  const int b   = by >> 8;

  uh[tid] = Uh[by * EE + tid];
  vv[tid] = Va[tid];
  __syncthreads();

  const int x = tid;
  const float4* wsr = (const float4*)(Ws + (b * TX + x) * EE);
  float acc = 0.0f;
#pragma unroll 4
  for (int e4 = 0; e4 < EE / 4; ++e4) {
    const float4 w = wsr[e4];
    const int e = e4 * 4;
    acc += fast_tanh(w.x + uh[e + 0]) * vv[e + 0];
    acc += fast_tanh(w.y + uh[e + 1]) * vv[e + 1];
    acc += fast_tanh(w.z + uh[e + 2]) * vv[e + 2];
    acc += fast_tanh(w.w + uh[e + 3]) * vv[e + 3];
  }

  // --- softmax over the 256 x-values held by the 256 threads ---
  red[tid] = acc;
  __syncthreads();
  for (int s = 128; s > 0; s >>= 1) {
    if (tid < s) red[tid] = fmaxf(red[tid], red[tid + s]);
    __syncthreads();
  }
  const float mx = red[0];
  __syncthreads();

  const float p = __expf(acc - mx);
  red[tid] = p;
  __syncthreads();
  for (int s = 128; s > 0; s >>= 1) {
    if (tid < s) red[tid] += red[tid + s];
    __syncthreads();
  }
  const float inv = 1.0f / red[0];

  e_out[by * TX + x] = p * inv;
}

// ---------------------------------------------------------------------------
// Stage 3: c[b] = e[b] @ enc[b]   (batched [256x256] @ [256x256]).
// ---------------------------------------------------------------------------
__global__ __launch_bounds__(128)
void ctx_kernel(const float* __restrict__ enc,
                const float* __restrict__ e_scores,
                float* __restrict__ c_out)
{
  const int wave = threadIdx.x >> 5;
  const int b  = blockIdx.z;
  const int m0 = (blockIdx.y * 4 + wave) * 16;
  const int n0 = blockIdx.x * 64;
  wave_gemm_tile_f32(e_scores + b * (TY * TX),
                     enc      + b * (TX * EE),
                     c_out    + b * (TY * EE),
                     m0, n0);
}

// ---------------------------------------------------------------------------
extern "C" void kernel_launch(void* const* d_in, const int* in_sizes, int n_in,
                              void* d_out, int out_size, void* d_ws, size_t ws_size,
                              hipStream_t stream)
{
  const float* enc = (const float*)d_in[0];   // [B,Tx,E]
  const float* dec = (const float*)d_in[1];   // [B,Ty,D]
  const float* Wa  = (const float*)d_in[2];   // [E,E]
  const float* Ua  = (const float*)d_in[3];   // [D,E]
  const float* Va  = (const float*)d_in[4];   // [E,1]

  float* c_out = (float*)d_out;               // [B,Ty,E]
  float* e_out = c_out + BSZ * TY * EE;       // [B,Ty,Tx]

  float* Ws = (float*)d_ws;                   // [B*Tx, E]  (2 MB)
  float* Uh = Ws + BSZ * TX * EE;             // [B*Ty, E]  (2 MB)

  // Stage 1: both input projections (z picks which GEMM).
  {
    dim3 blk(128, 1, 1);
    dim3 grd(EE / 64, (BSZ * TX) / 64, 2);    // 4 x 32 x 2
    proj_kernel<<<grd, blk, 0, stream>>>(enc, dec, Wa, Ua, Ws, Uh);
  }

  // Stage 2: fused tanh-dot + softmax -> e.
  score_kernel<<<BSZ * TY, 256, 0, stream>>>(Ws, Uh, Va, e_out);

  // Stage 3: context c = e @ enc.
  {
    dim3 blk(128, 1, 1);
    dim3 grd(EE / 64, TY / 64, BSZ);          // 4 x 4 x 8
    ctx_kernel<<<grd, blk, 0, stream>>>(enc, e_out, c_out);
  }
}